// RegionLoss_3D_multi_61263413510185
// MI455X (gfx1250) — compile-verified
//
#include <hip/hip_runtime.h>
#include <hip/hip_bf16.h>

typedef float v2f __attribute__((ext_vector_type(2)));
typedef float v8f __attribute__((ext_vector_type(8)));

#define NTOK 1152          // 1024 (8^3 tokens) + 128 (4^3 tokens)
#define CCH  64            // channels / K dim
#define NT16 72            // NTOK / 16

// ---------------------------------------------------------------------------
// Kernel 1: fused adaptive_avg_pool3d(8) + derived adaptive_avg_pool3d(4),
// written directly in token layout T[row][c] with row = concat(aa_rows, cc_rows).
// One block per (input, b, c); 512 threads, one 8x8x8 region per thread.
// Coalesced: wave reads contiguous 32B chunks (pw*8 floats) across lanes.
// ---------------------------------------------------------------------------
__global__ __launch_bounds__(512) void pool_tokens_kernel(
    const float* __restrict__ p1, const float* __restrict__ p2,
    float* __restrict__ t1, float* __restrict__ t2)
{
    const int blk = blockIdx.x;          // 0..255
    const int inp = blk >> 7;            // 0: p1, 1: p2
    const int b   = (blk >> 6) & 1;
    const int c   = blk & 63;

    const float* __restrict__ src = (inp == 0 ? p1 : p2);
    float* __restrict__ tok = (inp == 0 ? t1 : t2);

    const int r  = threadIdx.x;          // region id 0..511 (pd*64+ph*8+pw)
    const int pd = r >> 6;
    const int ph = (r >> 3) & 7;
    const int pw = r & 7;

    const size_t bc_off = ((size_t)(b * 64 + c)) << 18;   // *64*64*64
    const float* base = src + bc_off + (size_t)(pd * 8) * 4096
                             + (size_t)(ph * 8) * 64 + (size_t)(pw * 8);

    float s = 0.f;
#pragma unroll
    for (int i = 0; i < 8; ++i) {
#pragma unroll
        for (int j = 0; j < 8; ++j) {
            const float4* p = (const float4*)(base + i * 4096 + j * 64);
            float4 x = p[0];
            float4 y = p[1];
            s += x.x + x.y + x.z + x.w + y.x + y.y + y.z + y.w;
        }
    }
    const float mean8 = s * (1.0f / 512.0f);

    // aa/bb token: row = b*512 + r, col = c
    tok[(size_t)(b * 512 + r) * CCH + c] = mean8;

    // derive 4^3 pool from 8^3 pool (16^3 kernel == mean of 2x2x2 8^3 cells)
    __shared__ float smem[512];
    smem[r] = mean8;
    __syncthreads();

    if (r < 64) {
        const int qd = r >> 4;
        const int qh = (r >> 2) & 3;
        const int qw = r & 3;
        float t = 0.f;
#pragma unroll
        for (int od = 0; od < 2; ++od)
#pragma unroll
            for (int oh = 0; oh < 2; ++oh)
#pragma unroll
                for (int ow = 0; ow < 2; ++ow)
                    t += smem[(2 * qd + od) * 64 + (2 * qh + oh) * 8 + (2 * qw + ow)];
        const float mean4 = t * (1.0f / 8.0f);
        // cc/dd token: row = 1024 + b*64 + r
        tok[(size_t)(1024 + b * 64 + r) * CCH + c] = mean4;
    }
}

// ---------------------------------------------------------------------------
// Kernel 2: row-normalize both token matrices in place (cosine prep),
// norm = max(||row||, 1e-8). One wave per row; also zeroes d_out.
// ---------------------------------------------------------------------------
__global__ __launch_bounds__(256) void normalize_kernel(
    float* __restrict__ t1, float* __restrict__ t2, float* __restrict__ out)
{
    if (blockIdx.x == 0 && threadIdx.x == 0) out[0] = 0.0f;

    const int wave  = (blockIdx.x * blockDim.x + threadIdx.x) >> 5;  // 0..2303
    const int lane  = threadIdx.x & 31;
    if (wave >= 2 * NTOK) return;

    float* __restrict__ T = (wave < NTOK) ? t1 : t2;
    const int row = (wave < NTOK) ? wave : (wave - NTOK);

    float a = T[(size_t)row * CCH + lane];
    float b = T[(size_t)row * CCH + 32 + lane];
    float ss = a * a + b * b;
#pragma unroll
    for (int m = 16; m >= 1; m >>= 1)
        ss += __shfl_xor(ss, m, 32);

    const float nrm = fmaxf(sqrtf(ss), 1e-8f);
    const float inv = 1.0f / nrm;
    T[(size_t)row * CCH + lane]      = a * inv;
    T[(size_t)row * CCH + 32 + lane] = b * inv;
}

// ---------------------------------------------------------------------------
// Kernel 3: fused Gram-difference MSE using V_WMMA_F32_16X16X4_F32.
// One wave per 16x16 tile of sim = Pn @ Pn^T; 72x72 tiles; K=64 -> 16 WMMAs
// per Gram per tile. Accumulate sum((ga-gb)^2)/N^2 via one atomic per wave.
//
// Operand layout (ISA 16x4 f32 A / 4x16 f32 B): lanes 0-15 carry K={0,1} in
// VGPR{0,1}, lanes 16-31 carry K={2,3}. For Gram, B[k][n] = Pn[tile_j*16+n][k],
// so both operands load identically from row-major Pn as an aligned float2.
// ---------------------------------------------------------------------------
__global__ __launch_bounds__(256) void gram_mse_kernel(
    const float* __restrict__ pn1, const float* __restrict__ pn2,
    float* __restrict__ out)
{
    const int waveId = (blockIdx.x * blockDim.x + threadIdx.x) >> 5; // 0..5183
    const int lane   = threadIdx.x & 31;

    const int ti = waveId / NT16;
    const int tj = waveId - ti * NT16;

    const int m    = lane & 15;
    const int koff = (lane >> 4) << 1;            // 0 for lanes 0-15, 2 for 16-31

    const size_t rowA = (size_t)(ti * 16 + m) * CCH;
    const size_t rowB = (size_t)(tj * 16 + m) * CCH;

    v8f ga = {};
    v8f gb = {};

#pragma unroll
    for (int kk = 0; kk < 16; ++kk) {
        const int col = kk * 4 + koff;            // even -> 8-byte aligned
        v2f a1 = *(const v2f*)(pn1 + rowA + col);
        v2f b1 = *(const v2f*)(pn1 + rowB + col);
        v2f a2 = *(const v2f*)(pn2 + rowA + col);
        v2f b2 = *(const v2f*)(pn2 + rowB + col);
        // (neg_a, A, neg_b, B, c_mod, C, reuse_a, reuse_b)
        ga = __builtin_amdgcn_wmma_f32_16x16x4_f32(false, a1, false, b1,
                                                   (short)0, ga, false, false);
        gb = __builtin_amdgcn_wmma_f32_16x16x4_f32(false, a2, false, b2,
                                                   (short)0, gb, false, false);
    }

    float local = 0.f;
#pragma unroll
    for (int j = 0; j < 8; ++j) {
        float d = ga[j] - gb[j];
        local += d * d;
    }
#pragma unroll
    for (int msk = 16; msk >= 1; msk >>= 1)
        local += __shfl_xor(local, msk, 32);

    if (lane == 0) {
        const float inv_nn = 1.0f / ((float)NTOK * (float)NTOK);
        atomicAdd(out, local * inv_nn);
    }
}

// ---------------------------------------------------------------------------
extern "C" void kernel_launch(void* const* d_in, const int* in_sizes, int n_in,
                              void* d_out, int out_size, void* d_ws, size_t ws_size,
                              hipStream_t stream) {
    (void)in_sizes; (void)n_in; (void)out_size; (void)ws_size;

    const float* p1 = (const float*)d_in[0];
    const float* p2 = (const float*)d_in[1];
    float* out = (float*)d_out;

    float* t1 = (float*)d_ws;                      // [1152 x 64]
    float* t2 = t1 + (size_t)NTOK * CCH;           // [1152 x 64]

    // 1) pooling -> token matrices (256 MiB single pass, bandwidth bound)
    pool_tokens_kernel<<<dim3(256), dim3(512), 0, stream>>>(p1, p2, t1, t2);

    // 2) row normalization (one wave per row) + zero accumulator
    //    2*1152 rows, 8 waves per 256-thread block -> 288 blocks
    normalize_kernel<<<dim3(288), dim3(256), 0, stream>>>(t1, t2, out);

    // 3) WMMA Gram-diff MSE: 72*72 = 5184 tiles, 8 waves/block -> 648 blocks
    gram_mse_kernel<<<dim3(648), dim3(256), 0, stream>>>(t1, t2, out);
}